// GCN_26139170964145
// MI455X (gfx1250) — compile-verified
//
#include <hip/hip_runtime.h>

typedef float v2f __attribute__((ext_vector_type(2)));
typedef float v8f __attribute__((ext_vector_type(8)));

#define DIM 128
#define XP 132   // padded LDS pitch for x tile (kills bank conflicts on stride-128 reads)
#define NEG_SLOPE 0.1f

// ---- Phase 1: zero the aggregation buffer (d_out) and init deg=1 (self loop) ----
__global__ void gcn_init_kernel(float* __restrict__ out, float* __restrict__ deg,
                                int N, long total) {
  long i = (long)blockIdx.x * blockDim.x + threadIdx.x;
  if (i < total) out[i] = 0.0f;
  if (i < N) deg[i] = 1.0f;
}

// ---- Phase 2: degree on destination index ----
__global__ void gcn_deg_kernel(const int* __restrict__ ei, float* __restrict__ deg, int E) {
  int e = blockIdx.x * blockDim.x + threadIdx.x;
  if (e < E) atomicAdd(&deg[ei[E + e]], 1.0f);
}

// ---- Phase 3: dinv = rsqrt(deg) in place (deg >= 1 always, self loops) ----
__global__ void gcn_dinv_kernel(float* __restrict__ deg, int N) {
  int i = blockIdx.x * blockDim.x + threadIdx.x;
  if (i < N) deg[i] = rsqrtf(deg[i]);
}

// ---- Phase 4: h = x @ W via V_WMMA_F32_16X16X4_F32 ----
// Block: 256 threads = 8 waves. Block owns 16 rows of x; wave w owns cols [16w,16w+16).
// W (128x128 f32 = 64KB) staged in LDS once per block; x tile 16x128 padded to pitch 132.
__global__ __launch_bounds__(256) void gcn_gemm_kernel(const float* __restrict__ x,
                                                       const float* __restrict__ W,
                                                       float* __restrict__ h, int N) {
  __shared__ float Ws[DIM * DIM];
  __shared__ float xs[16 * XP];

  const int tid  = threadIdx.x;
  const int row0 = blockIdx.x * 16;

  // Cooperative load of W (float4, fully coalesced)
  for (int i = tid; i < (DIM * DIM) / 4; i += 256) {
    ((float4*)Ws)[i] = ((const float4*)W)[i];
  }
  // Cooperative load of x tile with padding (guard tail rows)
  for (int i = tid; i < 16 * DIM; i += 256) {
    int r = i >> 7, c = i & (DIM - 1);
    int gr = row0 + r;
    xs[r * XP + c] = (gr < N) ? x[(long)gr * DIM + c] : 0.0f;
  }
  __syncthreads();

  const int wave = tid >> 5;
  const int lane = tid & 31;
  const int half = lane >> 4;   // 0: lanes 0-15, 1: lanes 16-31
  const int m    = lane & 15;   // M index (A) / N index (B,C)
  const int col0 = wave * 16;

  v8f acc = {};
  // 32 WMMA ops: K=128 in steps of 4
  for (int k0 = 0; k0 < DIM; k0 += 4) {
    // A 16x4 f32 layout (ISA 7.12.2): lanes 0-15 -> K={k0,k0+1}, lanes 16-31 -> K={k0+2,k0+3}
    int ka = k0 + 2 * half;
    v2f a, b;
    a.x = xs[m * XP + ka];
    a.y = xs[m * XP + ka + 1];
    // B 4x16 f32: row striped across lanes within a VGPR; half-wave splits K the same way
    b.x = Ws[ka * DIM + col0 + m];
    b.y = Ws[(ka + 1) * DIM + col0 + m];
    acc = __builtin_amdgcn_wmma_f32_16x16x4_f32(
        /*neg_a=*/false, a, /*neg_b=*/false, b,
        /*c_mod=*/(short)0, acc, /*reuse_a=*/false, /*reuse_b=*/false);
  }

  // C/D 16x16 f32 layout: VGPR r -> M = r (lanes 0-15) / r+8 (lanes 16-31); N = lane&15
  for (int r = 0; r < 8; ++r) {
    int row = row0 + r + 8 * half;
    if (row < N) h[(long)row * DIM + col0 + m] = acc[r];
  }
}

// ---- Phase 5: edge scatter. One wave32 per edge, one float4 per lane (32*4 = 128 = D). ----
__global__ __launch_bounds__(256) void gcn_scatter_kernel(const int* __restrict__ ei,
                                                          const float* __restrict__ dinv,
                                                          const float* __restrict__ h,
                                                          float* __restrict__ out, int E) {
  int gwave = (int)((blockIdx.x * 256 + threadIdx.x) >> 5);
  int lane  = threadIdx.x & 31;
  if (gwave >= E) return;
  int s = ei[gwave];
  int d = ei[E + gwave];
  float w = dinv[s] * dinv[d];
  float4 v = ((const float4*)(h + (long)s * DIM))[lane];
  float* o = out + (long)d * DIM + lane * 4;
  atomicAdd(o + 0, v.x * w);
  atomicAdd(o + 1, v.y * w);
  atomicAdd(o + 2, v.z * w);
  atomicAdd(o + 3, v.w * w);
}

// ---- Phase 6: out = leaky(out + h*dinv^2 + b) + x (self-loop folded in) ----
__global__ void gcn_final_kernel(const float* __restrict__ x, const float* __restrict__ bias,
                                 const float* __restrict__ dinv, const float* __restrict__ h,
                                 float* __restrict__ out, long total) {
  long i = (long)blockIdx.x * blockDim.x + threadIdx.x;
  if (i >= total) return;
  int row = (int)(i >> 7);
  int col = (int)(i & (DIM - 1));
  float di = dinv[row];
  float v = out[i] + h[i] * (di * di) + bias[col];
  v = (v >= 0.0f) ? v : NEG_SLOPE * v;
  out[i] = v + x[i];
}

extern "C" void kernel_launch(void* const* d_in, const int* in_sizes, int n_in,
                              void* d_out, int out_size, void* d_ws, size_t ws_size,
                              hipStream_t stream) {
  const float* x  = (const float*)d_in[0];
  const int*   ei = (const int*)d_in[1];
  const float* W  = (const float*)d_in[2];
  const float* b  = (const float*)d_in[3];

  const int N = in_sizes[0] / DIM;   // 100000
  const int E = in_sizes[1] / 2;     // 1600000

  float* out  = (float*)d_out;
  float* dinv = (float*)d_ws;        // N floats: deg -> dinv in place
  size_t hoff = (((size_t)N * sizeof(float)) + 255) & ~(size_t)255;
  float* h    = (float*)((char*)d_ws + hoff);  // N*DIM floats

  const long total = (long)N * DIM;

  gcn_init_kernel<<<(int)((total + 255) / 256), 256, 0, stream>>>(out, dinv, N, total);
  gcn_deg_kernel<<<(E + 255) / 256, 256, 0, stream>>>(ei, dinv, E);
  gcn_dinv_kernel<<<(N + 255) / 256, 256, 0, stream>>>(dinv, N);
  gcn_gemm_kernel<<<(N + 15) / 16, 256, 0, stream>>>(x, W, h, N);
  gcn_scatter_kernel<<<(E + 7) / 8, 256, 0, stream>>>(ei, dinv, h, out, E);
  gcn_final_kernel<<<(int)((total + 255) / 256), 256, 0, stream>>>(x, b, dinv, h, out, total);
}